// GCN_52828097740997
// MI455X (gfx1250) — compile-verified
//
#include <hip/hip_runtime.h>
#include <math.h>

#define N_NODES   50000
#define N_EDGES   800000
#define NFEAT     128
#define NHID      128
#define NCLASS    10
#define N_GRAPHS  256
#define NTILES    (N_NODES / 16)   // 3125, exact

typedef __attribute__((ext_vector_type(2))) float v2f;
typedef __attribute__((ext_vector_type(8))) float v8f;

// Guaranteed single-instruction HW float atomic (no CAS-loop fallback).
// Non-returning -> tracked by STOREcnt; RMW executed at device scope (L2).
// S_ENDPGM performs an implicit wait-idle, so completion before kernel end
// is guaranteed without explicit s_wait_storecnt.
__device__ __forceinline__ void atomic_add_f32(float* p, float v) {
    asm volatile("global_atomic_add_f32 %0, %1, off scope:SCOPE_DEV"
                 :: "v"(p), "v"(v)
                 : "memory");
}

// ---------------------------------------------------------------- fill zero
__global__ void __launch_bounds__(256)
k_fill_zero(float* __restrict__ p, long n) {
    long i = (long)blockIdx.x * blockDim.x + threadIdx.x;
    if (i < n) p[i] = 0.0f;
}

// ------------------------------------------------- edge scatter (sum+count)
// one wave32 per edge; each lane moves float4 (32*4 = 128 channels)
__global__ void __launch_bounds__(256)
k_edge_scatter(const float* __restrict__ feat, const int* __restrict__ src,
               const int* __restrict__ dst, float* __restrict__ agg,
               int* __restrict__ cnt) {
    int tid  = blockIdx.x * blockDim.x + threadIdx.x;
    int e    = tid >> 5;
    int lane = tid & 31;
    if (e >= N_EDGES) return;
    int s = src[e];
    int d = dst[e];
    const float4* row = (const float4*)(feat + (long)s * NFEAT);
    float4 v = row[lane];
    float* out = agg + (long)d * NFEAT + lane * 4;
    atomic_add_f32(out + 0, v.x);
    atomic_add_f32(out + 1, v.y);
    atomic_add_f32(out + 2, v.z);
    atomic_add_f32(out + 3, v.w);
    if (lane == 0) atomicAdd(cnt + d, 1);
}

// ------------------------------------------------- divide by clamped count
__global__ void __launch_bounds__(256)
k_seg_normalize(float* __restrict__ acc, const int* __restrict__ cnt, int nrows) {
    int tid  = blockIdx.x * blockDim.x + threadIdx.x;
    int row  = tid >> 5;
    int lane = tid & 31;
    if (row >= nrows) return;
    int c = cnt[row];
    float inv = 1.0f / (float)(c > 0 ? c : 1);
    float4* p = (float4*)(acc + (long)row * NFEAT);
    float4 v = p[lane];
    v.x *= inv; v.y *= inv; v.z *= inv; v.w *= inv;
    p[lane] = v;
}

// ---------------- fused dual GEMM: out = relu(agg@Wn^T + b + x@Ws^T) -------
// one wave per 16-row tile; computes full 128 output cols (8 WMMA acc tiles)
__global__ void __launch_bounds__(256)
k_gcn_gemm_relu(const float* __restrict__ agg, const float* __restrict__ xin,
                const float* __restrict__ Wn, const float* __restrict__ Ws,
                const float* __restrict__ bias, float* __restrict__ out) {
    int wave = threadIdx.x >> 5;
    int lane = threadIdx.x & 31;
    int tile = blockIdx.x * 8 + wave;
    if (tile >= NTILES) return;   // wave-uniform: EXEC stays all-ones for WMMA

    int m  = lane & 15;            // row within tile (A) / col within tile (B)
    int kb = (lane >> 4) << 1;     // 0 or 2 : lanes 16-31 hold K+2,K+3
    long row0 = (long)tile * 16;

    v8f acc[8];
#pragma unroll
    for (int t = 0; t < 8; ++t) acc[t] = (v8f)(0.0f);

    const float* aBase = agg + (row0 + m) * NFEAT + kb;
    const float* xBase = xin + (row0 + m) * NFEAT + kb;

    // pass 1: agg @ Wn^T    (B[k][n] = Wn[n][k])
    for (int k0 = 0; k0 < NFEAT; k0 += 4) {
        v2f a = *(const v2f*)(aBase + k0);
#pragma unroll
        for (int t = 0; t < 8; ++t) {
            v2f b = *(const v2f*)(Wn + (long)(t * 16 + m) * NFEAT + k0 + kb);
            acc[t] = __builtin_amdgcn_wmma_f32_16x16x4_f32(
                false, a, false, b, (short)0, acc[t], false, false);
        }
    }
    // pass 2: x @ Ws^T accumulated into same C
    for (int k0 = 0; k0 < NFEAT; k0 += 4) {
        v2f a = *(const v2f*)(xBase + k0);
#pragma unroll
        for (int t = 0; t < 8; ++t) {
            v2f b = *(const v2f*)(Ws + (long)(t * 16 + m) * NFEAT + k0 + kb);
            acc[t] = __builtin_amdgcn_wmma_f32_16x16x4_f32(
                false, a, false, b, (short)0, acc[t], false, false);
        }
    }

    // epilogue: C/D layout -> row = r + (lane>=16 ? 8 : 0), col = lane&15
    int rowHi = (lane >> 4) * 8;
#pragma unroll
    for (int t = 0; t < 8; ++t) {
        int col = t * 16 + m;
        float bv = bias[col];
#pragma unroll
        for (int r = 0; r < 8; ++r) {
            float v = acc[t][r] + bv;
            v = v > 0.0f ? v : 0.0f;
            out[(row0 + rowHi + r) * NFEAT + col] = v;
        }
    }
}

// ------------------------------------------------- global mean pool scatter
__global__ void __launch_bounds__(256)
k_pool_scatter(const float* __restrict__ h, const int* __restrict__ batch,
               float* __restrict__ g, int* __restrict__ gcnt) {
    int tid  = blockIdx.x * blockDim.x + threadIdx.x;
    int node = tid >> 5;
    int lane = tid & 31;
    if (node >= N_NODES) return;
    int b = batch[node];
    const float4* row = (const float4*)(h + (long)node * NFEAT);
    float4 v = row[lane];
    float* out = g + (long)b * NFEAT + lane * 4;
    atomic_add_f32(out + 0, v.x);
    atomic_add_f32(out + 1, v.y);
    atomic_add_f32(out + 2, v.z);
    atomic_add_f32(out + 3, v.w);
    if (lane == 0) atomicAdd(gcnt + b, 1);
}

// -------------------------------- final linear + log_softmax, one block/graph
__global__ void __launch_bounds__(32)
k_final(const float* __restrict__ g, const float* __restrict__ Wlin,
        const float* __restrict__ blin, float* __restrict__ out) {
    int gid  = blockIdx.x;
    int lane = threadIdx.x;
    __shared__ float sl[16];
    if (lane < NCLASS) {
        const float* grow = g + (long)gid * NHID;
        const float* w = Wlin + (long)lane * NHID;
        float s = blin[lane];
        for (int k = 0; k < NHID; ++k) s += grow[k] * w[k];
        sl[lane] = s;
    }
    __syncthreads();
    if (lane == 0) {
        float mx = sl[0];
        for (int c = 1; c < NCLASS; ++c) mx = sl[c] > mx ? sl[c] : mx;
        float se = 0.0f;
        for (int c = 0; c < NCLASS; ++c) se += __expf(sl[c] - mx);
        float lse = mx + __logf(se);
        for (int c = 0; c < NCLASS; ++c) out[(long)gid * NCLASS + c] = sl[c] - lse;
    }
}

// ---------------------------------------------------------------------------
extern "C" void kernel_launch(void* const* d_in, const int* in_sizes, int n_in,
                              void* d_out, int out_size, void* d_ws, size_t ws_size,
                              hipStream_t stream) {
    const float* x    = (const float*)d_in[0];
    const int*   edge = (const int*)  d_in[1];
    const int*   batch= (const int*)  d_in[2];
    const float* W1n  = (const float*)d_in[3];
    const float* b1   = (const float*)d_in[4];
    const float* W1s  = (const float*)d_in[5];
    const float* W2n  = (const float*)d_in[6];
    const float* b2   = (const float*)d_in[7];
    const float* W2s  = (const float*)d_in[8];
    const float* W3n  = (const float*)d_in[9];
    const float* b3   = (const float*)d_in[10];
    const float* W3s  = (const float*)d_in[11];
    const float* Wlin = (const float*)d_in[12];
    const float* blin = (const float*)d_in[13];

    const int* src = edge;
    const int* dst = edge + N_EDGES;

    // workspace layout (agg|cnt contiguous for single zero-fill; g|gcnt too)
    const long FEAT_BYTES = (long)N_NODES * NFEAT * sizeof(float);   // 25.6 MB
    const long CNT_BYTES  = (long)N_NODES * sizeof(int);
    char* ws = (char*)d_ws;
    long off = 0;
    float* agg  = (float*)(ws + off); off += FEAT_BYTES;
    int*   cnt  = (int*)  (ws + off); off += CNT_BYTES;
    off = (off + 255) & ~255L;
    float* hA   = (float*)(ws + off); off += FEAT_BYTES;
    float* hB   = (float*)(ws + off); off += FEAT_BYTES;
    float* g    = (float*)(ws + off); off += (long)N_GRAPHS * NHID * sizeof(float);
    int*   gcnt = (int*)  (ws + off); off += (long)N_GRAPHS * sizeof(int);

    const long aggcnt_words = (long)N_NODES * NFEAT + N_NODES;
    const int  zeroBlocks   = (int)((aggcnt_words + 255) / 256);
    const int  edgeBlocks   = (N_EDGES * 32) / 256;                 // 100000
    const int  nodeW_blocks = (N_NODES * 32 + 255) / 256;           // 6250
    const int  gemmBlocks   = (NTILES + 7) / 8;                     // 391

    auto layer = [&](const float* fin, const float* Wn, const float* bs,
                     const float* Wss, float* hout) {
        k_fill_zero<<<zeroBlocks, 256, 0, stream>>>(agg, aggcnt_words);
        k_edge_scatter<<<edgeBlocks, 256, 0, stream>>>(fin, src, dst, agg, cnt);
        k_seg_normalize<<<nodeW_blocks, 256, 0, stream>>>(agg, cnt, N_NODES);
        k_gcn_gemm_relu<<<gemmBlocks, 256, 0, stream>>>(agg, fin, Wn, Wss, bs, hout);
    };

    layer(x,  W1n, b1, W1s, hA);
    layer(hA, W2n, b2, W2s, hB);
    layer(hB, W3n, b3, W3s, hA);

    const long gword = (long)N_GRAPHS * NHID + N_GRAPHS;
    k_fill_zero<<<(int)((gword + 255) / 256), 256, 0, stream>>>(g, gword);
    k_pool_scatter<<<nodeW_blocks, 256, 0, stream>>>(hA, batch, g, gcnt);
    k_seg_normalize<<<(N_GRAPHS * 32) / 256, 256, 0, stream>>>(g, gcnt, N_GRAPHS);
    k_final<<<N_GRAPHS, 32, 0, stream>>>(g, Wlin, blin, (float*)d_out);
}